// MPS_79937931313516
// MI455X (gfx1250) — compile-verified
//
#include <hip/hip_runtime.h>

#define D   512
#define DD  (D * D)
#define NSITES 62   // L-2 bulk sites

typedef __attribute__((ext_vector_type(2))) float v2f;
typedef __attribute__((ext_vector_type(8))) float v8f;

// D = A(16x4) * B(4x16) + C   (f32 WMMA, wave32)
__device__ __forceinline__ v8f wmma_f32(v2f a, v2f b, v8f c) {
  return __builtin_amdgcn_wmma_f32_16x16x4_f32(
      /*neg_a=*/false, a, /*neg_b=*/false, b,
      /*c_mod=*/(short)0, c, /*reuse_a=*/false, /*reuse_b=*/false);
}

// Row-contiguous fragment load: lane-half picks K-pair (rows row0, row0+1),
// l16 picks the column. Works for both A (M=l16) and B (N=l16) operands.
__device__ __forceinline__ v2f load_frag(const float* __restrict__ base,
                                         int row0, int col) {
  v2f f;
  f.x = base[row0 * D + col];
  f.y = base[(row0 + 1) * D + col];
  return f;
}

// cont0[j,k] = sum_s left[s,0,j] * conj(left[s,0,k])
__global__ __launch_bounds__(256)
void mps_init(const float* __restrict__ lr, const float* __restrict__ li,
              float* __restrict__ accr, float* __restrict__ acci) {
  int idx = blockIdx.x * 256 + threadIdx.x;    // 0..DD-1
  int j = idx >> 9;
  int k = idx & (D - 1);
  float r = 0.f, im = 0.f;
#pragma unroll
  for (int s = 0; s < 2; ++s) {
    float aj = lr[s * D + j], bj = li[s * D + j];
    float ak = lr[s * D + k], bk = li[s * D + k];
    r  += aj * ak + bj * bk;
    im += bj * ak - aj * bk;
  }
  accr[idx] = r;
  acci[idx] = im;
}

// Phase 1: t[s,k,j] = sum_i A[s,i,k] * acc[i,j]   (complex, A NOT conjugated)
// GEMM with Aop = A^T (row-contiguous frags), Bop = acc.
// 1x2 N-blocking: A fragments reused for two N tiles. Stores t TRANSPOSED.
__global__ __launch_bounds__(256)
void mps_phase1(const float* __restrict__ bulk_r, const float* __restrict__ bulk_i,
                const float* __restrict__ accr,  const float* __restrict__ acci,
                float* __restrict__ t_r, float* __restrict__ t_i, int site) {
  const int s    = blockIdx.z;
  const int wave = threadIdx.x >> 5;
  const int lane = threadIdx.x & 31;
  const int half = lane >> 4;
  const int l16  = lane & 15;
  const int gt   = blockIdx.x * 8 + wave;   // 0..511 wave-tiles
  const int KM   = (gt >> 4) << 4;          // 32 M tiles (k rows of t)
  const int JN   = (gt & 15) << 5;          // 16 N-pairs, 32 cols wide (j)

  const float* Ar = bulk_r + (size_t)(site * 2 + s) * DD;
  const float* Ai = bulk_i + (size_t)(site * 2 + s) * DD;

  v8f tr0 = {}, ti0 = {}, tr1 = {}, ti1 = {};
  const int rB  = 2 * half;
  const int cA  = KM + l16;
  const int cB0 = JN + l16;
  const int cB1 = JN + 16 + l16;
  for (int kk = 0; kk < D; kk += 4) {
    v2f ar  = load_frag(Ar,   kk + rB, cA);
    v2f ai  = load_frag(Ai,   kk + rB, cA);
    v2f nai = -ai;
    v2f br0 = load_frag(accr, kk + rB, cB0);
    v2f bi0 = load_frag(acci, kk + rB, cB0);
    v2f br1 = load_frag(accr, kk + rB, cB1);
    v2f bi1 = load_frag(acci, kk + rB, cB1);
    // t_r = Ar^T*acc_r - Ai^T*acc_i ; t_i = Ar^T*acc_i + Ai^T*acc_r
    tr0 = wmma_f32(ar,  br0, tr0);  tr0 = wmma_f32(nai, bi0, tr0);
    ti0 = wmma_f32(ar,  bi0, ti0);  ti0 = wmma_f32(ai,  br0, ti0);
    tr1 = wmma_f32(ar,  br1, tr1);  tr1 = wmma_f32(nai, bi1, tr1);
    ti1 = wmma_f32(ar,  bi1, ti1);  ti1 = wmma_f32(ai,  br1, ti1);
  }
  // store transposed: tT[j][k] = t[k][j]  (strided, once per 128 k-iters)
  float* dr = t_r + (size_t)s * DD;
  float* di = t_i + (size_t)s * DD;
  const int colb = KM + 8 * half;
  const int row0 = JN + l16;
  const int row1 = JN + 16 + l16;
#pragma unroll
  for (int v = 0; v < 8; ++v) {
    dr[row0 * D + colb + v] = tr0[v];
    di[row0 * D + colb + v] = ti0[v];
    dr[row1 * D + colb + v] = tr1[v];
    di[row1 * D + colb + v] = ti1[v];
  }
}

// Phase 2 (split-K over s): p[s][k,l] = sum_j t[s,k,j] * conj(A[s,j,l])
// Aop = tT rows (row-contiguous), Bop = A rows (row-contiguous).
// 1x2 N-blocking, coalesced stores into per-s partial planes.
__global__ __launch_bounds__(256)
void mps_phase2(const float* __restrict__ bulk_r, const float* __restrict__ bulk_i,
                const float* __restrict__ t_r, const float* __restrict__ t_i,
                float* __restrict__ p_r, float* __restrict__ p_i, int site) {
  const int s    = blockIdx.z;
  const int wave = threadIdx.x >> 5;
  const int lane = threadIdx.x & 31;
  const int half = lane >> 4;
  const int l16  = lane & 15;
  const int gt   = blockIdx.x * 8 + wave;   // 0..511
  const int KM   = (gt >> 4) << 4;          // rows of acc'
  const int LN   = (gt & 15) << 5;          // col pair base of acc'

  const float* tTr = t_r + (size_t)s * DD;
  const float* tTi = t_i + (size_t)s * DD;
  const float* Ar  = bulk_r + (size_t)(site * 2 + s) * DD;
  const float* Ai  = bulk_i + (size_t)(site * 2 + s) * DD;

  v8f cr0 = {}, ci0 = {}, cr1 = {}, ci1 = {};
  const int rB = 2 * half;
  const int cA  = KM + l16;
  const int cB0 = LN + l16;
  const int cB1 = LN + 16 + l16;
  for (int jj = 0; jj < D; jj += 4) {
    v2f atr  = load_frag(tTr, jj + rB, cA);
    v2f ati  = load_frag(tTi, jj + rB, cA);
    v2f natr = -atr;
    v2f bar0 = load_frag(Ar, jj + rB, cB0);
    v2f bai0 = load_frag(Ai, jj + rB, cB0);
    v2f bar1 = load_frag(Ar, jj + rB, cB1);
    v2f bai1 = load_frag(Ai, jj + rB, cB1);
    // p_r = t_r*Ar + t_i*Ai ; p_i = t_i*Ar - t_r*Ai   (conj on A)
    cr0 = wmma_f32(atr,  bar0, cr0);  cr0 = wmma_f32(ati, bai0, cr0);
    ci0 = wmma_f32(ati,  bar0, ci0);  ci0 = wmma_f32(natr, bai0, ci0);
    cr1 = wmma_f32(atr,  bar1, cr1);  cr1 = wmma_f32(ati, bai1, cr1);
    ci1 = wmma_f32(ati,  bar1, ci1);  ci1 = wmma_f32(natr, bai1, ci1);
  }
  float* outr = p_r + (size_t)s * DD;
  float* outi = p_i + (size_t)s * DD;
  const int row = KM + 8 * half;
#pragma unroll
  for (int v = 0; v < 8; ++v) {
    outr[(row + v) * D + LN + l16]      = cr0[v];
    outi[(row + v) * D + LN + l16]      = ci0[v];
    outr[(row + v) * D + LN + 16 + l16] = cr1[v];
    outi[(row + v) * D + LN + 16 + l16] = ci1[v];
  }
}

// acc = p[0] + p[1]  (both real and imag planes)
__global__ __launch_bounds__(256)
void mps_combine(const float* __restrict__ p_r, const float* __restrict__ p_i,
                 float* __restrict__ accr, float* __restrict__ acci) {
  int idx = blockIdx.x * 256 + threadIdx.x;   // 0..DD-1
  accr[idx] = p_r[idx] + p_r[DD + idx];
  acci[idx] = p_i[idx] + p_i[DD + idx];
}

// Right boundary: out = Re( sum_s u^T * acc * conj(u) ), u = right[s,:,0]
__global__ __launch_bounds__(512)
void mps_final(const float* __restrict__ rr, const float* __restrict__ ri,
               const float* __restrict__ accr, const float* __restrict__ acci,
               float* __restrict__ out) {
  __shared__ float red[512];
  const int j = threadIdx.x;   // 0..511
  float total = 0.f;
  for (int s = 0; s < 2; ++s) {
    const float* ur = rr + s * D;
    const float* ui = ri + s * D;
    float wr = 0.f, wi = 0.f;
    for (int i = 0; i < D; ++i) {
      float a = accr[i * D + j];   // coalesced across threads
      float b = acci[i * D + j];
      float x = ur[i], y = ui[i];
      wr += x * a - y * b;
      wi += x * b + y * a;
    }
    total += wr * ur[j] + wi * ui[j];
  }
  red[j] = total;
  __syncthreads();
  for (int off = 256; off > 0; off >>= 1) {
    if (j < off) red[j] += red[j + off];
    __syncthreads();
  }
  if (j == 0) out[0] = red[0];
}

extern "C" void kernel_launch(void* const* d_in, const int* in_sizes, int n_in,
                              void* d_out, int out_size, void* d_ws, size_t ws_size,
                              hipStream_t stream) {
  const float* left_r  = (const float*)d_in[0];
  const float* left_i  = (const float*)d_in[1];
  const float* bulk_r  = (const float*)d_in[2];
  const float* bulk_i  = (const float*)d_in[3];
  const float* right_r = (const float*)d_in[4];
  const float* right_i = (const float*)d_in[5];
  float* out = (float*)d_out;

  float* ws   = (float*)d_ws;
  float* accr = ws;            // DD
  float* acci = ws + DD;       // DD
  float* t_r  = ws + 2 * DD;   // 2*DD (s = 0,1)
  float* t_i  = ws + 4 * DD;   // 2*DD
  float* p_r  = ws + 6 * DD;   // 2*DD (per-s partials)
  float* p_i  = ws + 8 * DD;   // 2*DD

  mps_init<<<DD / 256, 256, 0, stream>>>(left_r, left_i, accr, acci);
  for (int site = 0; site < NSITES; ++site) {
    mps_phase1<<<dim3(64, 1, 2), 256, 0, stream>>>(bulk_r, bulk_i, accr, acci,
                                                   t_r, t_i, site);
    mps_phase2<<<dim3(64, 1, 2), 256, 0, stream>>>(bulk_r, bulk_i, t_r, t_i,
                                                   p_r, p_i, site);
    mps_combine<<<DD / 256, 256, 0, stream>>>(p_r, p_i, accr, acci);
  }
  mps_final<<<1, 512, 0, stream>>>(right_r, right_i, accr, acci, out);
}